// BackFlowTransformation_61237643706684
// MI455X (gfx1250) — compile-verified
//
#include <hip/hip_runtime.h>

typedef float v2f __attribute__((ext_vector_type(2)));
typedef float v8f __attribute__((ext_vector_type(8)));

#define NELEC 64
#define NDIM  3

// q_i = r_i * (1 + s_i) - (BF r)_i
//   BF_ij = w / r_ij (0 on diagonal), s_i = sum_j BF_ij
// Phase 1: d2 = A*B + C with A = R (16x4, K pad 0), B = -2*R^T, C = n2_i + n2_j + eye
// Phase 2: M = BF * [R | 1 | 0...]  (64x64 * 64x16, K=64 -> 16 chained WMMAs)
__global__ __launch_bounds__(128)
void backflow_wmma_kernel(const float* __restrict__ pos,
                          const float* __restrict__ wptr,
                          float* __restrict__ out)
{
    __shared__ float R4[NELEC][4];    // cols 0-2 = r, col 3 = 0 (K padding)
    __shared__ float n2[NELEC];       // |r_i|^2
    __shared__ float Bm[NELEC][16];   // cols 0-2 = r, col 3 = 1, cols 4-15 = 0
    __shared__ float BF[NELEC][68];   // backflow matrix, padded stride (bank spread)
    __shared__ float Mo[NELEC][4];    // M = BF*[R|1] columns 0..3

    const int b     = blockIdx.x;
    const int t     = threadIdx.x;        // 0..127
    const int wave  = t >> 5;             // 0..3 : 16-row strip owner
    const int lane  = t & 31;
    const int l16   = lane & 15;          // A-row / B-col / D-col within tile
    const int khalf = lane >> 4;          // selects K pair {0,1} or {2,3}
    const int row0  = wave * 16;

    const float w = wptr[0];
    const float* rp = pos + (size_t)b * (NELEC * NDIM);

    // ---- load R, build padded operands in LDS ----
    for (int i = t; i < NELEC; i += 128) {
        float x = rp[i * 3 + 0];
        float y = rp[i * 3 + 1];
        float z = rp[i * 3 + 2];
        R4[i][0] = x; R4[i][1] = y; R4[i][2] = z; R4[i][3] = 0.0f;
        n2[i] = x * x + y * y + z * z;
        Bm[i][0] = x; Bm[i][1] = y; Bm[i][2] = z; Bm[i][3] = 1.0f;
        #pragma unroll
        for (int c = 4; c < 16; ++c) Bm[i][c] = 0.0f;
    }
    __syncthreads();

    // ---- phase 1: Gram -> d2 -> BF (this wave's 16x64 strip) ----
    // A fragment: rows [row0, row0+16), K in {0,1} (lanes 0-15) or {2,3} (lanes 16-31)
    v2f a;
    a.x = R4[row0 + l16][2 * khalf + 0];
    a.y = R4[row0 + l16][2 * khalf + 1];

    for (int cb = 0; cb < 4; ++cb) {
        const int c0 = cb * 16;
        // B fragment: B[k][n] = -2 * R[c0+n][k]
        v2f bn;
        bn.x = -2.0f * R4[c0 + l16][2 * khalf + 0];
        bn.y = -2.0f * R4[c0 + l16][2 * khalf + 1];
        // C fragment preloads n2_i + n2_j + eye so D = d2 directly
        v8f cf;
        #pragma unroll
        for (int v = 0; v < 8; ++v) {
            const int gi = row0 + v + 8 * khalf;
            const int gj = c0 + l16;
            cf[v] = n2[gi] + n2[gj] + ((gi == gj) ? 1.0f : 0.0f);
        }
        v8f d2 = __builtin_amdgcn_wmma_f32_16x16x4_f32(
            false, a, false, bn, (short)0, cf, false, false);
        // bf = w * rsq(d2), diagonal masked to 0; raw v_rsq_f32 (no libm
        // denorm-guard branches: d2 is O(1) physical distances, diag forced 1)
        #pragma unroll
        for (int v = 0; v < 8; ++v) {
            const int gi = row0 + v + 8 * khalf;
            const int gj = c0 + l16;
            const float r = w * __builtin_amdgcn_rsqf(d2[v]);
            BF[gi][gj] = (gi == gj) ? 0.0f : r;
        }
    }
    __syncthreads();

    // ---- phase 2: M = BF * Bm  (rows [row0,row0+16), 16 cols, K=64) ----
    v8f acc = {};
    #pragma unroll
    for (int kb = 0; kb < 16; ++kb) {
        const int k0 = kb * 4 + 2 * khalf;
        v2f af, bf;
        af.x = BF[row0 + l16][k0 + 0];
        af.y = BF[row0 + l16][k0 + 1];
        bf.x = Bm[k0 + 0][l16];
        bf.y = Bm[k0 + 1][l16];
        acc = __builtin_amdgcn_wmma_f32_16x16x4_f32(
            false, af, false, bf, (short)0, acc, false, false);
    }
    if (l16 < 4) {
        #pragma unroll
        for (int v = 0; v < 8; ++v)
            Mo[row0 + v + 8 * khalf][l16] = acc[v];
    }
    __syncthreads();

    // ---- final combine: q[i][d] = r[i][d]*(1 + s_i) - M[i][d] ----
    for (int idx = t; idx < NELEC * NDIM; idx += 128) {
        const int i = idx / 3;
        const int d = idx % 3;
        out[(size_t)b * (NELEC * NDIM) + idx] =
            R4[i][d] * (1.0f + Mo[i][3]) - Mo[i][d];
    }
}

extern "C" void kernel_launch(void* const* d_in, const int* in_sizes, int n_in,
                              void* d_out, int out_size, void* d_ws, size_t ws_size,
                              hipStream_t stream) {
    const float* pos = (const float*)d_in[0];
    const float* w   = (const float*)d_in[1];
    float* out       = (float*)d_out;
    const int B = in_sizes[0] / (NELEC * NDIM);   // 4096
    backflow_wmma_kernel<<<B, 128, 0, stream>>>(pos, w, out);
}